// CRF_47493748359420
// MI455X (gfx1250) — compile-verified
//
#include <hip/hip_runtime.h>

// CRF loss (logZ - gold path score) for T=2048, L=1024, f32.
//
// Strategy (MI455X/gfx1250):
//  * Precompute E[i][j] = exp(transfer[i][j] - colmax[j]) once (4 MB, L2-resident).
//  * The sequential forward scan then becomes, per step, an exact f32 matvec
//    s[j] = sum_i exp(alpha[i]-c) * E[i][j], executed with V_WMMA_F32_16X16X4_F32
//    (rows of A replicated with the scan vector; all D rows identical).
//  * Persistent kernel: 16 WGs x 1024 threads, each WG owns 64 columns and keeps
//    its 1024x64 E slab in LDS (~266 KB of the 320 KB/WGP budget, stride-65 padded
//    for conflict-free B fetches). Device-wide step sync = software barrier with
//    agent-scope atomics + s_sleep. Deterministic (no FP atomics).

typedef __attribute__((ext_vector_type(2))) float v2f;
typedef __attribute__((ext_vector_type(8))) float v8f;

#define LBL 1024      // number of labels L
#define NWG 16        // persistent workgroups (one per WGP)
#define NT 1024       // threads per workgroup (32 waves)
#define CPW 64        // columns per workgroup  (NWG*CPW == LBL)
#define ESTRIDE 65    // LDS row stride (padding -> bank-conflict-free)

// ---------------------------------------------------------------- prep kernels

__global__ void crf_colmax_init(const float* __restrict__ transfer,
                                const float* __restrict__ feats,
                                float* __restrict__ colmax,
                                float* __restrict__ alpha) {
  int j = blockIdx.x * blockDim.x + threadIdx.x;
  if (j >= LBL) return;
  float m = -INFINITY;
  for (int i = 0; i < LBL; ++i) m = fmaxf(m, transfer[(size_t)i * LBL + j]);
  colmax[j] = m;
  alpha[j] = feats[j];            // alpha_0 = feats[0, :]
}

__global__ void crf_expm(const float* __restrict__ transfer,
                         const float* __restrict__ colmax,
                         float* __restrict__ E) {
  int idx = blockIdx.x * blockDim.x + threadIdx.x;   // exactly L*L threads
  float cm = colmax[idx & (LBL - 1)];
  E[idx] = __expf(transfer[idx] - cm);
}

__global__ void crf_gold_init(const float* __restrict__ feats,
                              const float* __restrict__ transfer,
                              const int* __restrict__ target,
                              float* __restrict__ gold,
                              int* __restrict__ bar, int T) {
  __shared__ float sred[256];
  int tid = threadIdx.x;
  float acc = 0.f;
  for (int t = tid; t < T; t += 256) {
    int lab = target[t];
    acc += feats[(size_t)t * LBL + lab];
    if (t > 0) acc += transfer[(size_t)target[t - 1] * LBL + lab];
  }
  sred[tid] = acc;
  __syncthreads();
  for (int s = 128; s > 0; s >>= 1) {
    if (tid < s) sred[tid] += sred[tid + s];
    __syncthreads();
  }
  if (tid == 0) {
    gold[0] = sred[0];
    bar[0] = 0;   // barrier arrival counter
    bar[1] = 0;   // barrier generation
  }
}

// ---------------------------------------------------------------- scan kernel

__device__ inline void grid_barrier(int* cnt, int* gen) {
  __syncthreads();
  if (threadIdx.x == 0) {
    int g = __hip_atomic_load(gen, __ATOMIC_RELAXED, __HIP_MEMORY_SCOPE_AGENT);
    __threadfence();
    int a = __hip_atomic_fetch_add(cnt, 1, __ATOMIC_ACQ_REL, __HIP_MEMORY_SCOPE_AGENT);
    if (a == NWG - 1) {
      __hip_atomic_store(cnt, 0, __ATOMIC_RELAXED, __HIP_MEMORY_SCOPE_AGENT);
      __hip_atomic_store(gen, g + 1, __ATOMIC_RELEASE, __HIP_MEMORY_SCOPE_AGENT);
    } else {
      while (__hip_atomic_load(gen, __ATOMIC_ACQUIRE, __HIP_MEMORY_SCOPE_AGENT) == g) {
        __builtin_amdgcn_s_sleep(2);
      }
    }
  }
  __syncthreads();
}

__global__ void __launch_bounds__(NT)
crf_scan(const float* __restrict__ feats,
         const float* __restrict__ Emat,
         const float* __restrict__ colmax,
         float* __restrict__ alpha,
         const float* __restrict__ gold,
         int* __restrict__ bar,
         float* __restrict__ out, int T) {
  extern __shared__ float smem[];
  float* E_lds  = smem;                      // LBL * ESTRIDE
  float* p_lds  = E_lds + LBL * ESTRIDE;     // LBL
  float* red    = p_lds + LBL;               // LBL (reductions)
  float* s_part = red + LBL;                 // 8 * CPW = 512 (K-slab partials)
  float* cm     = s_part + 8 * CPW;          // CPW

  const int tid  = threadIdx.x;
  const int lane = tid & 31;
  const int wave = tid >> 5;
  const int h    = lane >> 4;      // half-wave (selects K pair per ISA A/B layout)
  const int n    = lane & 15;
  const int wg   = blockIdx.x;
  const int j0   = wg * CPW;       // first owned column
  const int g    = wave & 3;       // column group (16 cols) within slab
  const int ks   = wave >> 2;      // K slab 0..7 (each 128 wide)
  const int k0   = ks * 128;
  const int cb   = g * 16 + n;     // column index within slab, 0..63

  // Stage this WGP's E slab (1024 x 64, padded) into LDS once.
  for (int idx = tid; idx < LBL * CPW; idx += NT) {
    int i  = idx >> 6;
    int jc = idx & 63;
    E_lds[i * ESTRIDE + jc] = Emat[(size_t)i * LBL + j0 + jc];
  }
  if (tid < CPW) cm[tid] = colmax[j0 + tid];
  __syncthreads();

  for (int t = 1; t < T; ++t) {
    // ---- global max of alpha (wave shuffles + 1 LDS hop) ----
    float av = alpha[tid];
    float m = av;
    for (int off = 16; off > 0; off >>= 1) m = fmaxf(m, __shfl_xor(m, off, 32));
    if (lane == 0) red[wave] = m;
    __syncthreads();
    if (tid < 32) {
      float mm = red[lane];
      for (int off = 16; off > 0; off >>= 1) mm = fmaxf(mm, __shfl_xor(mm, off, 32));
      if (lane == 0) red[0] = mm;
    }
    __syncthreads();
    const float c = red[0];
    p_lds[tid] = __expf(av - c);
    __syncthreads();

    if (t + 1 < T) __builtin_prefetch(feats + (size_t)(t + 1) * LBL + j0, 0, 1);

    // ---- matvec via V_WMMA_F32_16X16X4_F32 over this wave's K slab ----
    // A layout (16x4 f32): lane l holds row l%16, k = 2*(l/16)+v  -> replicate p.
    // B layout (4x16 f32): lane l holds col l%16, k = 2*(l/16)+v.
    v8f acc = {};
#pragma unroll 4
    for (int kk = 0; kk < 128; kk += 4) {
      const int kb = k0 + kk + 2 * h;
      v2f A, B;
      A.x = p_lds[kb];
      A.y = p_lds[kb + 1];
      B.x = E_lds[kb * ESTRIDE + cb];
      B.y = E_lds[(kb + 1) * ESTRIDE + cb];
      acc = __builtin_amdgcn_wmma_f32_16x16x4_f32(false, A, false, B,
                                                  (short)0, acc, false, false);
    }
    // All D rows identical; VGPR0 lane l holds s for column g*16 + (l%16).
    if (h == 0) s_part[ks * CPW + cb] = acc[0];
    __syncthreads();

    // ---- combine K slabs, finish logsumexp, publish alpha_t ----
    if (tid < CPW) {
      float s = 0.f;
#pragma unroll
      for (int q = 0; q < 8; ++q) s += s_part[q * CPW + tid];
      float a_new = c + cm[tid] + __logf(s) + feats[(size_t)t * LBL + j0 + tid];
      alpha[j0 + tid] = a_new;
    }
    grid_barrier(bar, bar + 1);
  }

  // ---- epilogue: logZ = logsumexp(alpha_final); out = logZ - gold ----
  if (wg == 0) {
    float av = alpha[tid];
    float m = av;
    for (int off = 16; off > 0; off >>= 1) m = fmaxf(m, __shfl_xor(m, off, 32));
    if (lane == 0) red[wave] = m;
    __syncthreads();
    if (tid < 32) {
      float mm = red[lane];
      for (int off = 16; off > 0; off >>= 1) mm = fmaxf(mm, __shfl_xor(mm, off, 32));
      if (lane == 0) red[0] = mm;
    }
    __syncthreads();
    const float c2 = red[0];
    float e = __expf(av - c2);
    for (int off = 16; off > 0; off >>= 1) e += __shfl_xor(e, off, 32);
    if (lane == 0) red[32 + wave] = e;
    __syncthreads();
    if (tid == 0) {
      float s = 0.f;
      for (int q = 0; q < 32; ++q) s += red[32 + q];
      out[0] = c2 + __logf(s) - gold[0];
    }
  }
}

// ---------------------------------------------------------------- launcher

extern "C" void kernel_launch(void* const* d_in, const int* in_sizes, int n_in,
                              void* d_out, int out_size, void* d_ws, size_t ws_size,
                              hipStream_t stream) {
  const float* feats    = (const float*)d_in[0];   // [T, L]
  const float* transfer = (const float*)d_in[1];   // [L, L]
  const int*   target   = (const int*)d_in[2];     // [T]
  const int T = in_sizes[2];                       // 2048

  // workspace layout (floats): E[L*L] | colmax[L] | alpha[L] | gold[1] | bar[2 ints]
  float* ws     = (float*)d_ws;
  float* E      = ws;
  float* colmax = ws + (size_t)LBL * LBL;
  float* alpha  = colmax + LBL;
  float* gold   = alpha + LBL;
  int*   bar    = (int*)(gold + 1);

  crf_colmax_init<<<dim3(LBL / 256), dim3(256), 0, stream>>>(transfer, feats, colmax, alpha);
  crf_expm<<<dim3((LBL * LBL) / 256), dim3(256), 0, stream>>>(transfer, colmax, E);
  crf_gold_init<<<dim3(1), dim3(256), 0, stream>>>(feats, transfer, target, gold, bar, T);

  const size_t smem_bytes =
      (size_t)(LBL * ESTRIDE + LBL + LBL + 8 * CPW + CPW) * sizeof(float);
  crf_scan<<<dim3(NWG), dim3(NT), smem_bytes, stream>>>(
      feats, E, colmax, alpha, gold, bar, (float*)d_out, T);
}